// GenomeNetTorch_81930796138998
// MI455X (gfx1250) — compile-verified
//
#include <hip/hip_runtime.h>

// Problem dims (fixed by the reference)
#define Bdim 1024
#define Ndim 4096
#define Kdim 16
#define Ldim 8

#define NSPLIT 8                 // n-range splits -> 64*8 = 512 workgroups per layer
#define NR (Ndim / NSPLIT)       // 512 rows of n per workgroup
#define LDS_STRIDE 17            // 16 floats per b-tile row + 1 pad (bank-conflict-free)
#define SLAB_BYTES (Ndim * LDS_STRIDE * 4)   // 278528 B <= 320 KB WGP LDS

typedef unsigned int v4u __attribute__((ext_vector_type(4)));
typedef int          v8i __attribute__((ext_vector_type(8)));
typedef int          v4i __attribute__((ext_vector_type(4)));

// Detect therock-10.0 HIP headers (amdgpu-toolchain) -> 6-arg TDM builtin,
// else ROCm 7.2 -> 5-arg TDM builtin.
#if defined(__has_include)
#if __has_include(<hip/amd_detail/amd_gfx1250_TDM.h>)
#define TDM_SIX_ARGS 1
#endif
#endif
#ifndef TDM_SIX_ARGS
#define TDM_SIX_ARGS 0
#endif

// tanh without any libm/device-lib call.
// gfx1250 has V_TANH_F32; clang exposes __builtin_amdgcn_tanhf. Fallback:
// tanh(x) = sign(x) * (1-e)/(1+e),  e = 2^(-2*log2(e)*|x|)   (overflow-safe)
__device__ __forceinline__ float fast_tanh(float x) {
#if __has_builtin(__builtin_amdgcn_tanhf)
  return __builtin_amdgcn_tanhf(x);
#else
  const float a = __builtin_fabsf(x);
  const float e = __builtin_amdgcn_exp2f(-2.8853900817779268f * a);
  const float t = (1.0f - e) * __builtin_amdgcn_rcpf(1.0f + e);
  return __builtin_copysignf(t, x);
#endif
}

// ---------------------------------------------------------------------------
// Prep: interleave (src, w) -> int2 pairs so the layer kernel fetches each
// k-row as 8x b128 broadcast loads (no shuffles needed).
// ---------------------------------------------------------------------------
__global__ __launch_bounds__(256) void pack_iw(
    const int* __restrict__ src, const float* __restrict__ w,
    int2* __restrict__ iw, int total)
{
  const int t = blockIdx.x * 256 + threadIdx.x;
  if (t < total) iw[t] = make_int2(src[t], __float_as_int(w[t]));
}

// ---------------------------------------------------------------------------
// Layer kernel: vout[n, b] = tanh( sum_k vin[src[n,k], b] * w[n,k] )
// vin/vout are (N, B) transposed activations. Each workgroup owns a 16-wide
// b-tile and an NR-long n-range. The whole b-slab vin[:, btile..btile+15]
// (4096 x 16 f32) is DMA'd into LDS by the Tensor Data Mover with 1-DWORD
// padding every 16 DWORDs (LDS row stride 17 -> conflict-free gathers).
// ---------------------------------------------------------------------------
__global__ __launch_bounds__(256) void genome_layer(
    const float* __restrict__ vin,   // (N, B)
    float*       __restrict__ vout,  // (N, B)
    const int2*  __restrict__ iwl)   // (N, K) packed (idx, w) for this layer
{
  extern __shared__ float slab[];    // Ndim * LDS_STRIDE floats
  const int tid   = threadIdx.x;
  const int btile = blockIdx.x << 4;        // 16 b's per workgroup
  const int nbase = blockIdx.y * NR;

  // ---- TDM: one tensor_load_to_lds for the whole 278KB slab (wave 0 only) ----
  if (tid == 0) {
    unsigned long long ga = (unsigned long long)(const void*)(vin + btile);
    unsigned int lds0 = __builtin_amdgcn_groupstaticsize();  // dynamic-LDS base

    v4u g0;
    g0[0] = 1u;                                            // count=1, user D#
    g0[1] = lds0;                                          // lds_addr (bytes)
    g0[2] = (unsigned int)(ga & 0xFFFFFFFFull);            // global_addr[31:0]
    g0[3] = (unsigned int)(((ga >> 32) & 0x01FFFFFFull)    // global_addr[56:32]
                           | 0x80000000u);                 // type=2 ("image")

    v8i g1;
    g1[0] = 0x00D20000;          // data_size=4B, pad_enable=1,
                                 // pad_interval=3 (16 DW), pad_amount=0 (1 DW)
    g1[1] = (int)(1024u << 16);  // tensor_dim0 = 1024 (b extent)
    g1[2] = (int)(4096u << 16);  // tensor_dim0 hi=0 | tensor_dim1 lo = 4096
    g1[3] = (int)(16u << 16);    // tensor_dim1 hi=0 | tile_dim0 = 16
    g1[4] = 4096;                // tile_dim1 = 4096 rows, tile_dim2 = 0 (2D)
    g1[5] = 1024;                // tensor_dim0_stride = B elements
    g1[6] = 0;                   // stride hi | tensor_dim1_stride lo (unused)
    g1[7] = 0;

    v4i gz = {0, 0, 0, 0};
#if TDM_SIX_ARGS
    v8i gz8 = {0, 0, 0, 0, 0, 0, 0, 0};
    __builtin_amdgcn_tensor_load_to_lds(g0, g1, gz, gz, gz8, 0);
#else
    __builtin_amdgcn_tensor_load_to_lds(g0, g1, gz, gz, 0);
#endif
    __builtin_amdgcn_s_wait_tensorcnt(0);
  }
  __syncthreads();   // slab visible to all 8 waves

  // ---- Gather + dot + tanh. Lane layout: bb = tid&15, two n's per wave. ----
  // Per 32 outputs: 8x global b128 (pair rows, L2-broadcast) + 16x ds_load
  // (conflict-free stride-17 gathers) + 16 FMA + v_tanh.
  const int bb = tid & 15;
  for (int n = nbase + (tid >> 4); n < nbase + NR; n += 16) {
    const int4* row = (const int4*)(iwl + (size_t)n * Kdim);  // 128B, 16B-aligned
    float acc = 0.0f;
#pragma unroll
    for (int j = 0; j < 8; ++j) {
      const int4 q = row[j];     // (idx0, w0, idx1, w1)
      acc = __builtin_fmaf(slab[q.x * LDS_STRIDE + bb], __int_as_float(q.y), acc);
      acc = __builtin_fmaf(slab[q.z * LDS_STRIDE + bb], __int_as_float(q.w), acc);
    }
    vout[(size_t)n * Bdim + btile + bb] = fast_tanh(acc);
  }
}

// ---------------------------------------------------------------------------
// Generic 32x32 LDS-tile transpose: out[c*R + r] = in[r*C + c]
// ---------------------------------------------------------------------------
__global__ __launch_bounds__(256) void transpose_k(
    const float* __restrict__ in, float* __restrict__ out, int R, int C)
{
  __shared__ float tile[32][33];
  const int cb = blockIdx.x * 32;
  const int rb = blockIdx.y * 32;
  const int tx = threadIdx.x;
  const int ty = threadIdx.y;
#pragma unroll
  for (int i = 0; i < 32; i += 8)
    tile[ty + i][tx] = in[(size_t)(rb + ty + i) * C + (cb + tx)];
  __syncthreads();
#pragma unroll
  for (int i = 0; i < 32; i += 8)
    out[(size_t)(cb + ty + i) * R + (rb + tx)] = tile[tx][ty + i];
}

// ---------------------------------------------------------------------------
extern "C" void kernel_launch(void* const* d_in, const int* in_sizes, int n_in,
                              void* d_out, int out_size, void* d_ws, size_t ws_size,
                              hipStream_t stream) {
  (void)in_sizes; (void)n_in; (void)out_size; (void)ws_size;
  const float* x   = (const float*)d_in[0];   // (B, N) f32
  const int*   src = (const int*)d_in[1];     // (L, N, K) i32
  const float* w   = (const float*)d_in[2];   // (L, N, K) f32
  float* out = (float*)d_out;                 // (B, N) f32

  float* bufA = (float*)d_ws;                       // (N, B) ping      16 MB
  float* bufB = bufA + (size_t)Ndim * Bdim;         // (N, B) pong      16 MB
  int2*  iw   = (int2*)(bufB + (size_t)Ndim * Bdim);// (L, N, K) pairs   4 MB

  // Pack (src, w) -> interleaved pairs (whole net, 512K elements)
  const int total = Ldim * Ndim * Kdim;
  pack_iw<<<dim3((total + 255) / 256), dim3(256), 0, stream>>>(src, w, iw, total);

  // x (B,N) -> bufA (N,B)
  transpose_k<<<dim3(Ndim / 32, Bdim / 32), dim3(32, 8), 0, stream>>>(
      x, bufA, Bdim, Ndim);

  const float* cur = bufA;
  float*       nxt = bufB;
  for (int l = 0; l < Ldim; ++l) {
    genome_layer<<<dim3(Bdim / 16, NSPLIT), dim3(256), SLAB_BYTES, stream>>>(
        cur, nxt, iw + (size_t)l * Ndim * Kdim);
    float* t = (float*)cur; cur = nxt; nxt = t;
  }

  // final activations (N,B) -> out (B,N)
  transpose_k<<<dim3(Bdim / 32, Ndim / 32), dim3(32, 8), 0, stream>>>(
      cur, out, Ndim, Bdim);
}